// Net0_36146444763843
// MI455X (gfx1250) — compile-verified
//
#include <hip/hip_runtime.h>

// ---------------------------------------------------------------------------
// 20-layer tiny MLP (H=7) + log_softmax on V_WMMA_F32_16X16X4_F32 (gfx1250).
//
// D = A(16x4 weights, rows = permuted output features) x B(4x16 activations,
// cols = 16 samples) + C.  Row permutation pi: M0->f0 M1->f1 M2->f4 M3->f5
// M8->f2 M9->f3 M10->f6 M11->f7(const 1).  With pi, WMMA output registers
// D0..D3 are bit-identical to the next layer's B operand (K rows 0/2, 1/3,
// 4/6, 5/7) -- zero shuffles between layers.  Bias folded into K=7 via the
// constant-1 feature-7 row.  4 independent sample tiles per wave iteration
// (64 samples) keep 4 WMMA chains in flight on the XDL pipe.
// ReLU done as integer smax (v_max_i32) to avoid the float canonicalize.
// ---------------------------------------------------------------------------

typedef __attribute__((ext_vector_type(2))) float v2f;
typedef __attribute__((ext_vector_type(8))) float v8f;

#define N_MID 19
#define HDIM  7
#define TILES 4                    // 16-sample tiles per wave iteration

__device__ __forceinline__ v8f wmma4(v2f a, v2f b, v8f c) {
  // (neg_a, A, neg_b, B, c_mod, C, reuse_a, reuse_b)
  return __builtin_amdgcn_wmma_f32_16x16x4_f32(false, a, false, b, (short)0, c,
                                               false, false);
}

// 1-instruction ReLU: for non-NaN floats, IEEE compare vs +0 == signed int
// compare, so relu(x) = int_as_float(smax(float_as_int(x), 0)) -> v_max_i32.
__device__ __forceinline__ float relu1(float v) {
  int i = __float_as_int(v);
  i = (i > 0) ? i : 0;
  return __int_as_float(i);
}
__device__ __forceinline__ void relu4(v8f& c) {
  c[0] = relu1(c[0]);
  c[1] = relu1(c[1]);
  c[2] = relu1(c[2]);
  c[3] = relu1(c[3]);   // M=11 ("ones" row, 0x3F800000) stays exactly 1.0
}

__global__ __launch_bounds__(256) void mlp20_wmma_kernel(
    const float* __restrict__ x,
    const float* __restrict__ W1,  const float* __restrict__ b1,
    const float* __restrict__ Wmid,const float* __restrict__ bmid,
    const float* __restrict__ W21, const float* __restrict__ b21,
    float* __restrict__ out, long long nSamples)
{
  const int lane = threadIdx.x & 31;
  const int hi   = lane >> 4;        // 0: lanes 0-15 (K 0,1 / 4,5), 1: lanes 16-31 (K 2,3 / 6,7)
  const int m    = lane & 15;        // A-matrix row M for this lane
  const int mm   = m & 7;
  // row permutation pi(m) -> feature index, or -1 for zero rows
  const int f = (mm < 4) ? ((mm < 2 ? mm : mm + 2) + ((m & 8) ? 2 : 0)) : -1;

  // ---- mid-layer weight fragments (per-lane gather, hits L2; done once) ----
  auto midW = [&](int layer, int k) -> float {   // A[m][k], k in 0..7
    if (f < 0)  return 0.f;
    if (f == 7) return (k == 7) ? 1.f : 0.f;     // ones-row
    if (k == 7) return bmid[layer * HDIM + f];   // bias folded at K=7
    return Wmid[(layer * HDIM + f) * HDIM + k];
  };

  v2f Alo[N_MID], Ahi[N_MID];
  #pragma unroll
  for (int i = 0; i < N_MID; ++i) {
    Alo[i].x = midW(i, 0 + 2 * hi);  // K = 0 / 2
    Alo[i].y = midW(i, 1 + 2 * hi);  // K = 1 / 3
    Ahi[i].x = midW(i, 4 + 2 * hi);  // K = 4 / 6
    Ahi[i].y = midW(i, 5 + 2 * hi);  // K = 5 / 7 (bias)
  }

  // ---- fc1 fragment: K=0,1 = inputs, K=2 = bias slot, K=3 = 0 ----
  v2f A1;
  if (f < 0)       { A1.x = 0.f;                     A1.y = 0.f; }
  else if (f == 7) { A1.x = hi ? 1.f : 0.f;          A1.y = 0.f; }
  else             { A1.x = hi ? b1[f] : W1[f*2+0];  A1.y = hi ? 0.f : W1[f*2+1]; }

  // ---- fc21 fragments: 2 output rows (M=0,1), bias at K=7 ----
  const int f2 = (m == 0) ? 0 : (m == 1) ? 1 : -1;
  auto w21 = [&](int k) -> float {
    if (f2 < 0) return 0.f;
    if (k == 7) return b21[f2];
    return W21[f2 * HDIM + k];
  };
  v2f A21lo, A21hi;
  A21lo.x = w21(0 + 2 * hi);  A21lo.y = w21(1 + 2 * hi);
  A21hi.x = w21(4 + 2 * hi);  A21hi.y = w21(5 + 2 * hi);

  const v8f z = {0.f, 0.f, 0.f, 0.f, 0.f, 0.f, 0.f, 0.f};

  const long long nWaves = (long long)gridDim.x * (blockDim.x >> 5);
  const long long gw = (((long long)blockIdx.x * blockDim.x) + threadIdx.x) >> 5;
  const long long step = 16 * TILES;          // samples per wave iteration
  const long long stride = nWaves * step;

  for (long long s = gw * step; s < nSamples; s += stride) {
    // speculative prefetch of next iteration's x (global_prefetch_b8)
    __builtin_prefetch((const void*)(x + (s + stride) * 2), 0, 1);

    // ---- load x, build fc1 B operands (K=0:x0, K=1:x1, K=2:1, K=3:0) ----
    v2f bx[TILES];
    if (!hi) {
      #pragma unroll
      for (int t = 0; t < TILES; ++t)
        bx[t] = *(const v2f*)(x + (s + 16 * t + m) * 2);
    } else {
      #pragma unroll
      for (int t = 0; t < TILES; ++t) { bx[t].x = 1.f; bx[t].y = 0.f; }
    }

    // ---- fc1 + relu ----
    v8f c[TILES];
    #pragma unroll
    for (int t = 0; t < TILES; ++t) c[t] = wmma4(A1, bx[t], z);
    #pragma unroll
    for (int t = 0; t < TILES; ++t) relu4(c[t]);

    // ---- fc2..fc20: 2 WMMAs/layer/tile, D regs reused directly as B ----
    #pragma unroll
    for (int i = 0; i < N_MID; ++i) {
      v2f bl[TILES], bh[TILES];
      #pragma unroll
      for (int t = 0; t < TILES; ++t) {
        bl[t].x = c[t][0]; bl[t].y = c[t][1];
        bh[t].x = c[t][2]; bh[t].y = c[t][3];
      }
      v8f n[TILES];
      #pragma unroll
      for (int t = 0; t < TILES; ++t) n[t] = wmma4(Alo[i], bl[t], z);
      #pragma unroll
      for (int t = 0; t < TILES; ++t) n[t] = wmma4(Ahi[i], bh[t], n[t]);
      #pragma unroll
      for (int t = 0; t < TILES; ++t) { relu4(n[t]); c[t] = n[t]; }
    }

    // ---- fc21 (logits in rows M=0,1) + 2-class log_softmax ----
    v2f o[TILES];
    #pragma unroll
    for (int t = 0; t < TILES; ++t) {
      v2f bl; bl.x = c[t][0]; bl.y = c[t][1];
      v2f bh; bh.x = c[t][2]; bh.y = c[t][3];
      v8f L = wmma4(A21lo, bl, z);
      L = wmma4(A21hi, bh, L);
      float l0 = L[0], l1 = L[1];
      float mx  = fmaxf(l0, l1);
      float lse = mx + __logf(__expf(l0 - mx) + __expf(l1 - mx));
      o[t].x = l0 - lse;
      o[t].y = l1 - lse;
    }

    if (!hi) {
      #pragma unroll
      for (int t = 0; t < TILES; ++t)
        *(v2f*)(out + (s + 16 * t + m) * 2) = o[t];
    }
  }
}

extern "C" void kernel_launch(void* const* d_in, const int* in_sizes, int n_in,
                              void* d_out, int out_size, void* d_ws, size_t ws_size,
                              hipStream_t stream) {
  const float* x    = (const float*)d_in[0];
  const float* W1   = (const float*)d_in[1];
  const float* b1   = (const float*)d_in[2];
  const float* Wmid = (const float*)d_in[3];
  const float* bmid = (const float*)d_in[4];
  const float* W21  = (const float*)d_in[5];
  const float* b21  = (const float*)d_in[6];
  float* out = (float*)d_out;

  const long long nSamples = (long long)in_sizes[0] / 2;  // x is [B,2]

  dim3 block(256);          // 8 waves per workgroup
  dim3 grid(1024);          // 8192 waves, grid-stride over B/64 iterations
  hipLaunchKernelGGL(mlp20_wmma_kernel, grid, block, 0, stream,
                     x, W1, b1, Wmid, bmid, W21, b21, out, nSamples);
}